// MessagePassingElectionModel_48378511622646
// MI455X (gfx1250) — compile-verified
//
#include <hip/hip_runtime.h>
#include <hip/hip_bf16.h>
#include <math.h>

// ---------------- problem constants (match reference) ----------------
#define N_NODES   50000
#define N_EDGES   1600000
#define N_CAND    1000
#define N_GRAPHS  50
#define EMB       32
#define L_LAYERS  4
#define EPSV      1e-5f

// Edge-kernel tiling: 128 threads = 4 waves; each wave owns 16 edges -> 64 edges/block.
#define EDGES_PER_BLOCK 64
#define BLK_THREADS     128
// 1600000 / 64 = 25000 blocks exactly (no tail).
#define EDGE_BLOCKS     (N_EDGES / EDGES_PER_BLOCK)

typedef float v8f __attribute__((ext_vector_type(8)));
typedef float v2f __attribute__((ext_vector_type(2)));

// D = A(16x4 f32) * B(4x16 f32) + C(16x16 f32), full-precision CDNA5 WMMA.
__device__ __forceinline__ v8f wmma_f32_k4(v2f a, v2f b, v8f c) {
    return __builtin_amdgcn_wmma_f32_16x16x4_f32(
        /*neg_a=*/false, a, /*neg_b=*/false, b,
        /*c_mod=*/(short)0, c, /*reuse_a=*/false, /*reuse_b=*/false);
}

// ---------------- kernel 1: h0 = x @ W_in + b_in ----------------
__global__ void k_input(const float* __restrict__ x,
                        const float* __restrict__ W_in,
                        const float* __restrict__ b_in,
                        float* __restrict__ h) {
    int i = blockIdx.x * blockDim.x + threadIdx.x;
    if (i >= N_NODES * EMB) return;
    int n = i >> 5;
    int c = i & 31;
    h[i] = fmaf(x[2 * n], W_in[c], fmaf(x[2 * n + 1], W_in[EMB + c], b_in[c]));
}

// ---------------- kernel 2: h_next = h  (float4 copy) ----------------
__global__ void k_copy4(const float4* __restrict__ in, float4* __restrict__ out, int n4) {
    int i = blockIdx.x * blockDim.x + threadIdx.x;
    if (i < n4) out[i] = in[i];
}

// ---------------- kernel 3: one message-passing layer ----------------
// msg = [h[dst] | h[src] | edge_attr] (65 -> pad 68)
// t = relu(bn1(msg @ W1)); t = relu(bn2(t @ W2)); h_next[dst] += t  (atomic)
__global__ __launch_bounds__(BLK_THREADS)
void k_edge_layer(const float* __restrict__ h_in,
                  float* __restrict__ h_next,
                  const int* __restrict__ ei,      // [2, E]
                  const float* __restrict__ ea,    // [E, 1]
                  const float* __restrict__ W1,    // [L,65,32]
                  const float* __restrict__ b1,
                  const float* __restrict__ g1,
                  const float* __restrict__ be1,
                  const float* __restrict__ m1,
                  const float* __restrict__ v1,
                  const float* __restrict__ W2,    // [L,32,32]
                  const float* __restrict__ b2,
                  const float* __restrict__ g2,
                  const float* __restrict__ be2,
                  const float* __restrict__ m2,
                  const float* __restrict__ v2,
                  int layer) {
    __shared__ float MSGs[EDGES_PER_BLOCK * 68];   // 64 x 68 msg rows (pad cols 65..67 = 0)
    __shared__ float W1t[32 * 72];                 // W1 transposed: [n][k], stride 72, k 65..67 = 0
    __shared__ float W2t[32 * 36];                 // W2 transposed: [n][k], stride 36
    __shared__ float Ts[4 * 16 * 36];              // per-wave 16x32 intermediate, stride 36
    __shared__ float A1s[32], B1s[32], A2s[32], B2s[32];
    __shared__ int   DSTs[EDGES_PER_BLOCK];

    const int tid = threadIdx.x;

    // ---- load + transpose weights into LDS; fold BN into per-channel (A,B) ----
    for (int i = tid; i < 32 * 68; i += BLK_THREADS) {
        int n = i / 68, k = i - n * 68;
        W1t[n * 72 + k] = (k < 65) ? W1[(layer * 65 + k) * 32 + n] : 0.0f;
    }
    for (int i = tid; i < 32 * 32; i += BLK_THREADS) {
        int n = i >> 5, k = i & 31;
        W2t[n * 36 + k] = W2[(layer * 32 + k) * 32 + n];
    }
    if (tid < 32) {
        int n = tid, o = layer * 32 + n;
        float s1 = g1[o] * rsqrtf(v1[o] + EPSV);
        A1s[n] = s1;
        B1s[n] = fmaf(b1[o] - m1[o], s1, be1[o]);
        float s2 = g2[o] * rsqrtf(v2[o] + EPSV);
        A2s[n] = s2;
        B2s[n] = fmaf(b2[o] - m2[o], s2, be2[o]);
    }

    // ---- gather phase: 2 threads per edge row ----
    {
        int r    = tid >> 1;
        int half = tid & 1;
        long e   = (long)blockIdx.x * EDGES_PER_BLOCK + r;
        int dst  = ei[N_EDGES + e];   // aggregation target, first in concat
        int src  = ei[e];
        float4*       mrow = (float4*)(MSGs + r * 68 + half * 32);
        const float4* hrow = (const float4*)(h_in + (size_t)(half ? src : dst) * EMB);
#pragma unroll
        for (int i = 0; i < 8; ++i) mrow[i] = hrow[i];
        if (half) {
            MSGs[r * 68 + 64] = ea[e];
            MSGs[r * 68 + 65] = 0.0f;
            MSGs[r * 68 + 66] = 0.0f;
            MSGs[r * 68 + 67] = 0.0f;
        } else {
            DSTs[r] = dst;
        }
    }
    __syncthreads();

    // ---- per-wave WMMA MLP over its 16 edges ----
    const int lane  = tid & 31;
    const int wave  = tid >> 5;
    const int ml    = lane & 15;            // M (A) / N (B,C) index within tile
    const int khalf = (lane >> 4) << 1;     // lanes 16..31 carry K+2 halves
    const int mrow0 = (lane >> 4) << 3;     // C layout: lanes 16..31 hold rows M+8

    const float* Arow = MSGs + (wave * 16 + ml) * 68;
    const float* Bn0  = W1t + ml * 72;
    const float* Bn1  = W1t + (16 + ml) * 72;

    v8f c0 = {}; v8f c1 = {};
#pragma unroll
    for (int kk = 0; kk < 17; ++kk) {        // K = 68 in steps of 4
        int kb = kk * 4 + khalf;
        v2f a;  a.x  = Arow[kb]; a.y  = Arow[kb + 1];
        v2f b0; b0.x = Bn0[kb];  b0.y = Bn0[kb + 1];
        v2f b1v; b1v.x = Bn1[kb]; b1v.y = Bn1[kb + 1];
        c0 = wmma_f32_k4(a, b0, c0);
        c1 = wmma_f32_k4(a, b1v, c1);
    }

    // BN1 + ReLU on accumulators (channel is lane-uniform), restage to LDS in A-layout
    float* Tw = Ts + wave * 16 * 36;
    {
        float sa0 = A1s[ml], sb0 = B1s[ml];
        float sa1 = A1s[16 + ml], sb1 = B1s[16 + ml];
#pragma unroll
        for (int r = 0; r < 8; ++r) {
            int m = mrow0 + r;
            Tw[m * 36 + ml]      = fmaxf(fmaf(c0[r], sa0, sb0), 0.0f);
            Tw[m * 36 + 16 + ml] = fmaxf(fmaf(c1[r], sa1, sb1), 0.0f);
        }
    }
    // DS ops are in-order within a wave: stores above complete before loads below.

    const float* A2row = Tw + ml * 36;
    const float* Cn0   = W2t + ml * 36;
    const float* Cn1   = W2t + (16 + ml) * 36;
    v8f d0 = {}; v8f d1 = {};
#pragma unroll
    for (int kk = 0; kk < 8; ++kk) {         // K = 32 in steps of 4
        int kb = kk * 4 + khalf;
        v2f a;  a.x  = A2row[kb]; a.y  = A2row[kb + 1];
        v2f b0; b0.x = Cn0[kb];   b0.y = Cn0[kb + 1];
        v2f b1v; b1v.x = Cn1[kb]; b1v.y = Cn1[kb + 1];
        d0 = wmma_f32_k4(a, b0, d0);
        d1 = wmma_f32_k4(a, b1v, d1);
    }

    // BN2 + ReLU + atomic scatter-add into h_next[dst]
    {
        float ua0 = A2s[ml], ub0 = B2s[ml];
        float ua1 = A2s[16 + ml], ub1 = B2s[16 + ml];
#pragma unroll
        for (int r = 0; r < 8; ++r) {
            int m   = mrow0 + r;
            int dst = DSTs[wave * 16 + m];
            float o0 = fmaxf(fmaf(d0[r], ua0, ub0), 0.0f);
            float o1 = fmaxf(fmaf(d1[r], ua1, ub1), 0.0f);
            unsafeAtomicAdd(h_next + (size_t)dst * EMB + ml,      o0);
            unsafeAtomicAdd(h_next + (size_t)dst * EMB + 16 + ml, o1);
        }
    }
}

// ---------------- kernel 4: candidate logits + segmented log-softmax ----------------
__global__ __launch_bounds__(256)
void k_readout(const float* __restrict__ h,
               const int* __restrict__ cand,
               const int* __restrict__ batch,
               const float* __restrict__ W_out,
               const float* __restrict__ b_out,
               float* __restrict__ out) {
    __shared__ float lg[N_CAND];
    __shared__ int   sg[N_CAND];
    __shared__ float gmax[N_GRAPHS], glse[N_GRAPHS];

    int tid = threadIdx.x;
    for (int c = tid; c < N_CAND; c += 256) {
        int node = cand[c];
        const float* hr = h + (size_t)node * EMB;
        float acc = b_out[0];
#pragma unroll
        for (int k = 0; k < EMB; ++k) acc = fmaf(hr[k], W_out[k], acc);
        lg[c] = acc;
        sg[c] = batch[node];
    }
    __syncthreads();
    for (int g = tid; g < N_GRAPHS; g += 256) {
        float m = -INFINITY;
        for (int c = 0; c < N_CAND; ++c)
            if (sg[c] == g) m = fmaxf(m, lg[c]);
        float s = 0.0f;
        for (int c = 0; c < N_CAND; ++c)
            if (sg[c] == g) s += __expf(lg[c] - m);
        gmax[g] = m;
        glse[g] = __logf(s);
    }
    __syncthreads();
    for (int c = tid; c < N_CAND; c += 256) {
        int g = sg[c];
        out[c] = lg[c] - gmax[g] - glse[g];
    }
}

// ---------------- launcher ----------------
extern "C" void kernel_launch(void* const* d_in, const int* in_sizes, int n_in,
                              void* d_out, int out_size, void* d_ws, size_t ws_size,
                              hipStream_t stream) {
    const float* x      = (const float*)d_in[0];
    const int*   ei     = (const int*)  d_in[1];
    const float* ea     = (const float*)d_in[2];
    const int*   cand   = (const int*)  d_in[3];
    const int*   batch  = (const int*)  d_in[4];
    const float* W_in   = (const float*)d_in[5];
    const float* b_in   = (const float*)d_in[6];
    const float* W1     = (const float*)d_in[7];
    const float* b1     = (const float*)d_in[8];
    const float* g1     = (const float*)d_in[9];
    const float* be1    = (const float*)d_in[10];
    const float* m1     = (const float*)d_in[11];
    const float* v1     = (const float*)d_in[12];
    const float* W2     = (const float*)d_in[13];
    const float* b2     = (const float*)d_in[14];
    const float* g2     = (const float*)d_in[15];
    const float* be2    = (const float*)d_in[16];
    const float* m2     = (const float*)d_in[17];
    const float* v2     = (const float*)d_in[18];
    const float* W_out  = (const float*)d_in[19];
    const float* b_out  = (const float*)d_in[20];
    float*       out    = (float*)d_out;

    // workspace: two ping-pong node-feature buffers (needs 2 * 6.4 MB)
    float* hA = (float*)d_ws;
    float* hB = hA + (size_t)N_NODES * EMB;

    const int NH  = N_NODES * EMB;      // 1,600,000 floats
    const int NH4 = NH / 4;

    // h0 = x @ W_in + b_in
    k_input<<<(NH + 255) / 256, 256, 0, stream>>>(x, W_in, b_in, hA);

    float* h_cur = hA;
    float* h_nxt = hB;
    for (int l = 0; l < L_LAYERS; ++l) {
        // h_next starts as h (residual), edges accumulate on top
        k_copy4<<<(NH4 + 255) / 256, 256, 0, stream>>>((const float4*)h_cur, (float4*)h_nxt, NH4);
        k_edge_layer<<<EDGE_BLOCKS, BLK_THREADS, 0, stream>>>(
            h_cur, h_nxt, ei, ea,
            W1, b1, g1, be1, m1, v1,
            W2, b2, g2, be2, m2, v2, l);
        float* t = h_cur; h_cur = h_nxt; h_nxt = t;
    }

    k_readout<<<1, 256, 0, stream>>>(h_cur, cand, batch, W_out, b_out, out);
    (void)in_sizes; (void)n_in; (void)out_size; (void)ws_size;
}